// ExpertLinearValve_69157563400656
// MI455X (gfx1250) — compile-verified
//
#include <hip/hip_runtime.h>
#include <math.h>

// Problem constants (from the reference): B=8, T=8192, E=1, R=64, K=16, I=256, O=256
#define NTOK   65536   // B*T*E rows of X
#define IFEAT  256
#define OFEAT  256
#define RANK   64
#define TOPK   16

typedef __attribute__((ext_vector_type(2))) float v2f;
typedef __attribute__((ext_vector_type(8))) float v8f;

// ---------------------------------------------------------------------------
// Kernel 1: column sums of X [NTOK x IFEAT] -> sum[IFEAT]  (sum must be 0-init)
// 256 blocks x 256 threads; thread t owns column t for 256 rows -> coalesced.
// ---------------------------------------------------------------------------
__global__ __launch_bounds__(256)
void colsum_kernel(const float* __restrict__ x, float* __restrict__ sum) {
    const int t = threadIdx.x;
    const size_t row0 = (size_t)blockIdx.x * 256;
    const float* p = x + row0 * IFEAT + t;
    float acc = 0.0f;
#pragma unroll 8
    for (int r = 0; r < 256; ++r) acc += p[(size_t)r * IFEAT];
    atomicAdd(&sum[t], acc);
}

// ---------------------------------------------------------------------------
// Kernel 2: predictor + softplus + top-16 via parallel rank.
// s[r] = coeff[r]^2 if rank(r) < 16 else 0, where
//   rank(r) = #{ j : coeff[j] > coeff[r] || (coeff[j]==coeff[r] && j < r) }
// This exactly matches jax.lax.top_k (ties -> lowest index) and avoids the
// serial scalar selection loop (which exploded into ~10k SALU instructions).
// ---------------------------------------------------------------------------
__global__ __launch_bounds__(256)
void predictor_kernel(const float* __restrict__ sum,
                      const float* __restrict__ pred_w,
                      const float* __restrict__ pred_b,
                      float* __restrict__ s) {
    __shared__ float xm[IFEAT];
    __shared__ float coeff[RANK];
    const int t = threadIdx.x;
    xm[t] = sum[t] * (1.0f / (float)NTOK);
    __syncthreads();
    if (t < RANK) {
        float z = pred_b[t];
        const float* wrow = pred_w + (size_t)t * IFEAT;
#pragma unroll 8
        for (int i = 0; i < IFEAT; ++i) z += xm[i] * wrow[i];
        // numerically stable softplus: log1p(exp(z))
        coeff[t] = (z > 20.0f) ? z : log1pf(expf(z));
    }
    __syncthreads();
    if (t < RANK) {
        const float c = coeff[t];
        int rank = 0;
#pragma unroll 8
        for (int j = 0; j < RANK; ++j) {
            const float cj = coeff[j];
            rank += (cj > c || (cj == c && j < t)) ? 1 : 0;
        }
        s[t] = (rank < TOPK) ? c * c : 0.0f;
    }
}

// ---------------------------------------------------------------------------
// Kernel 3: W[o,i] = sum_r s[r] * down[o,r] * up[r,i]
// 256 blocks (o) x 256 threads (i).
// ---------------------------------------------------------------------------
__global__ __launch_bounds__(256)
void build_weight_kernel(const float* __restrict__ up,
                         const float* __restrict__ down,
                         const float* __restrict__ s,
                         float* __restrict__ W) {
    __shared__ float sd[RANK];   // s[r] * down[o,r]
    const int o = blockIdx.x;
    const int i = threadIdx.x;
    if (i < RANK) sd[i] = s[i] * down[(size_t)o * RANK + i];
    __syncthreads();
    float acc = 0.0f;
#pragma unroll 8
    for (int r = 0; r < RANK; ++r) acc += sd[r] * up[(size_t)r * IFEAT + i];
    W[(size_t)o * IFEAT + i] = acc;
}

// ---------------------------------------------------------------------------
// Kernel 4: out = X @ W^T + bias using V_WMMA_F32_16X16X4_F32 (wave32).
// Block = 256 threads = 8 waves. Block covers 32 rows x 256 cols:
//   wave w: m_tile = w>>2 (0..1), n_group = w&3 (each n_group = 64 cols = 4 tiles)
// Each wave: 4 x v8f accumulators, K loop 256 in steps of 4 -> 256 WMMAs/wave,
// with 4x in-register reuse of the A operand per iteration.
//
// ISA VGPR layouts (CDNA5 ISA):
//   A 16x4 f32 : lane L holds M=L%16,  VGPR j = K = (L>>4)*2 + j     -> float2
//   B 4x16 f32 : lane L holds N=L%16,  VGPR j = K = (L>>4)*2 + j; B[k][n]=W[n][k]
//   C/D 16x16  : component v at M = v + 8*(L>>4), N = L%16
// ---------------------------------------------------------------------------
__global__ __launch_bounds__(256)
void gemm_wmma_kernel(const float* __restrict__ X,
                      const float* __restrict__ W,
                      const float* __restrict__ bias,
                      float* __restrict__ out) {
    const int lane  = threadIdx.x & 31;
    const int wave  = threadIdx.x >> 5;
    const int mt    = wave >> 2;        // 0..1  (16-row tile within block)
    const int ng    = wave & 3;         // 0..3  (64-col group)
    const int l16   = lane & 15;
    const int khalf = (lane >> 4) * 2;  // 0 or 2

    const int arow = blockIdx.x * 32 + mt * 16 + l16;       // X row for A operand
    const float* xrow = X + (size_t)arow * IFEAT + khalf;

    const int nbase = ng * 64;
    const float* wrow0 = W + (size_t)(nbase +  0 + l16) * IFEAT + khalf;
    const float* wrow1 = W + (size_t)(nbase + 16 + l16) * IFEAT + khalf;
    const float* wrow2 = W + (size_t)(nbase + 32 + l16) * IFEAT + khalf;
    const float* wrow3 = W + (size_t)(nbase + 48 + l16) * IFEAT + khalf;

    v8f acc0, acc1, acc2, acc3;
    {
        const float b0 = bias[nbase +  0 + l16];
        const float b1 = bias[nbase + 16 + l16];
        const float b2 = bias[nbase + 32 + l16];
        const float b3 = bias[nbase + 48 + l16];
#pragma unroll
        for (int v = 0; v < 8; ++v) { acc0[v] = b0; acc1[v] = b1; acc2[v] = b2; acc3[v] = b3; }
    }

#pragma unroll 4
    for (int kb = 0; kb < IFEAT; kb += 4) {
        v2f a  = *(const v2f*)(xrow  + kb);
        v2f b0 = *(const v2f*)(wrow0 + kb);
        v2f b1 = *(const v2f*)(wrow1 + kb);
        v2f b2 = *(const v2f*)(wrow2 + kb);
        v2f b3 = *(const v2f*)(wrow3 + kb);
        acc0 = __builtin_amdgcn_wmma_f32_16x16x4_f32(false, a, false, b0, (short)0, acc0, false, false);
        acc1 = __builtin_amdgcn_wmma_f32_16x16x4_f32(false, a, false, b1, (short)0, acc1, false, false);
        acc2 = __builtin_amdgcn_wmma_f32_16x16x4_f32(false, a, false, b2, (short)0, acc2, false, false);
        acc3 = __builtin_amdgcn_wmma_f32_16x16x4_f32(false, a, false, b3, (short)0, acc3, false, false);
    }

    // Store D: component v -> row m = v + 8*(lane>>4), col n = tile base + l16
    const int mrow0 = blockIdx.x * 32 + mt * 16 + 8 * (lane >> 4);
#pragma unroll
    for (int v = 0; v < 8; ++v) {
        const size_t rbase = (size_t)(mrow0 + v) * OFEAT + nbase + l16;
        out[rbase +  0] = acc0[v];
        out[rbase + 16] = acc1[v];
        out[rbase + 32] = acc2[v];
        out[rbase + 48] = acc3[v];
    }
}

// ---------------------------------------------------------------------------
// Launcher
// ---------------------------------------------------------------------------
extern "C" void kernel_launch(void* const* d_in, const int* in_sizes, int n_in,
                              void* d_out, int out_size, void* d_ws, size_t ws_size,
                              hipStream_t stream) {
    const float* x      = (const float*)d_in[0];  // [8,8192,1,256]
    const float* up     = (const float*)d_in[1];  // [1,64,256]
    const float* down   = (const float*)d_in[2];  // [1,256,64]
    const float* bias   = (const float*)d_in[3];  // [1,256]
    const float* pred_w = (const float*)d_in[4];  // [64,256]
    const float* pred_b = (const float*)d_in[5];  // [64]
    // d_in[6] = k (always 16 here)

    float* ws    = (float*)d_ws;
    float* wsum  = ws;            // 256 floats (needs zero init for atomics)
    float* wsel  = ws + 256;      // 64 floats: s[r]
    float* wmat  = ws + 512;      // 65536 floats: W[o,i]
    float* out   = (float*)d_out;

    hipMemsetAsync(wsum, 0, IFEAT * sizeof(float), stream);
    colsum_kernel<<<256, 256, 0, stream>>>(x, wsum);
    predictor_kernel<<<1, 256, 0, stream>>>(wsum, pred_w, pred_b, wsel);
    build_weight_kernel<<<OFEAT, IFEAT, 0, stream>>>(up, down, wsel, wmat);
    gemm_wmma_kernel<<<NTOK / 32, 256, 0, stream>>>(x, wmat, bias, out);
}